// CTCLayer_29549374996919
// MI455X (gfx1250) — compile-verified
//
#include <hip/hip_runtime.h>

// CTC forward (negative log-likelihood), blank = C-1.
// y_true [256,128] int32, y_pred [256,1024,256] f32, out [256] f32.

#define B_   256
#define T_   1024
#define C_   256
#define L_   128
#define S_   257        // 2*L + 1
#define NBUF 8          // frame ring depth (power of 2)
#define NEGV (-1e30f)
#define EPSV (1e-7f)

// Async DMA of 16 bytes/lane from global memory into LDS (CDNA5 async path).
__device__ __forceinline__ void async_load_b128(void* lds_dst, const void* gsrc) {
    unsigned int       l = (unsigned int)(unsigned long long)lds_dst;
    unsigned long long g = (unsigned long long)gsrc;
    asm volatile("global_load_async_to_lds_b128 %0, %1, off"
                 :: "v"(l), "v"(g)
                 : "memory");
}

__device__ __forceinline__ void wait_async_le5() { asm volatile("s_wait_asynccnt 0x5" ::: "memory"); }
__device__ __forceinline__ void wait_async_le0() { asm volatile("s_wait_asynccnt 0x0" ::: "memory"); }

// Split workgroup barrier (CDNA5): flush LDS ops, then signal; wait consumes completion.
__device__ __forceinline__ void bar_signal() {
    asm volatile("s_wait_dscnt 0x0\n\ts_barrier_signal -1" ::: "memory");
}
__device__ __forceinline__ void bar_wait() {
    asm volatile("s_barrier_wait -1" ::: "memory");
}

__launch_bounds__(288, 1)
__global__ void ctc_forward_kernel(const int*   __restrict__ labels,   // [B_, L_]
                                   const float* __restrict__ y_pred,   // [B_, T_, C_]
                                   float*       __restrict__ out) {    // [B_]
    __shared__ float frame[NBUF][C_];    // 8-deep ring of per-frame prob rows (8 KB)
    __shared__ float alpha[2][S_ + 2];   // [0..1] = NEG pads, state s at index s+2

    const int  b      = blockIdx.x;
    const int  tid    = threadIdx.x;
    const bool active = (tid < S_);
    const bool issuer = (tid < 64);      // waves 0&1: 64 lanes x 16B = 1024B/frame

    // Per-state constants: extended label + skip-transition mask.
    int  myExt  = C_ - 1;   // blank (even s)
    bool mySkip = false;
    if (active && (tid & 1)) {
        const int k = (tid - 1) >> 1;
        myExt = labels[b * L_ + k];
        if (tid >= 3) mySkip = (myExt != labels[b * L_ + k - 1]);
    }

    // NEG pads so s-1 / s-2 reads are branch-free at the left edge.
    if (tid < 2) { alpha[0][tid] = NEGV; alpha[1][tid] = NEGV; }

    const float* base  = y_pred + (size_t)b * (size_t)(T_ * C_);
    const float* gnext = base + tid * 4;        // issuer lane: next frame, 16B chunk
    float*       lbase = &frame[0][tid * 4];    // issuer lane: slot 0 chunk

    // Prologue: fill the ring with frames 0..NBUF-1; confirm frames 0..2.
    if (issuer) {
#pragma unroll
        for (int f = 0; f < NBUF; ++f) {
            async_load_b128(lbase + f * C_, gnext);
            gnext += C_;
        }
        wait_async_le5();      // 8 in flight -> 3 oldest (frames 0,1,2) landed
    }
    bar_signal();              // round A: frame residency published to all 9 waves
    bar_wait();

    // ---- t = 0: init, and prefetch lp for t = 1 ----
    if (active) {
        alpha[0][tid + 2] = (tid < 2) ? __logf(frame[0][myExt] + EPSV) : NEGV;
    }
    bar_signal();              // round 0: alpha(0) published
    float lp_pre = 0.0f;
    if (active) lp_pre = __logf(frame[1][myExt] + EPSV);   // off critical path

    // ---- main recurrence: split barrier per step, lp prefetched a step ahead ----
    for (int t = 1; t < T_; ++t) {
        bar_wait();            // consume round t-1: alpha(t-1) + frame t+1 resident

        // Refill the slot frame t-1 vacated (last read in iter t-2) with frame t+7.
        if (issuer && t <= T_ - NBUF) {
            async_load_b128(lbase + ((t - 1) & (NBUF - 1)) * C_, gnext);
            gnext += C_;
        }

        if (active) {
            const float* ap = alpha[(t - 1) & 1];
            const float  a  = ap[tid + 2];
            const float  s1 = ap[tid + 1];
            const float  s2 = mySkip ? ap[tid] : NEGV;
            const float  m  = fmaxf(a, fmaxf(s1, s2));
            const float  sm = __expf(a - m) + __expf(s1 - m) + __expf(s2 - m);
            alpha[t & 1][tid + 2] = lp_pre + m + __logf(sm);
        }

        // Maintain invariant: frame t+2 confirmed before signaling round t.
        if (issuer) {
            if      (t <= T_ - NBUF) wait_async_le5();
            else if (t <= T_ - 3)    wait_async_le0();
        }
        bar_signal();          // round t: alpha(t) + frame t+2 residency published

        // Prefetch lp(t+1) behind the signal (hidden from the critical chain).
        if (active && t <= T_ - 2) {
            lp_pre = __logf(frame[(t + 1) & (NBUF - 1)][myExt] + EPSV);
        }
    }

    bar_wait();                // consume round T-1: alpha(T-1) published
    if (tid == 0) {
        const float* ap = alpha[(T_ - 1) & 1];
        const float x = ap[(S_ - 1) + 2];
        const float y = ap[(S_ - 2) + 2];
        const float m = fmaxf(x, y);
        out[b] = -(m + __logf(__expf(x - m) + __expf(y - m)));
    }
}

extern "C" void kernel_launch(void* const* d_in, const int* in_sizes, int n_in,
                              void* d_out, int out_size, void* d_ws, size_t ws_size,
                              hipStream_t stream) {
    (void)in_sizes; (void)n_in; (void)d_ws; (void)ws_size; (void)out_size;
    const int*   y_true = (const int*)d_in[0];
    const float* y_pred = (const float*)d_in[1];
    float*       out    = (float*)d_out;
    ctc_forward_kernel<<<B_, 288, 0, stream>>>(y_true, y_pred, out);
}